// QSRGAN_81320910782763
// MI455X (gfx1250) — compile-verified
//
#include <hip/hip_runtime.h>

// QSRGAN quantum-generator head for MI455X (gfx1250).
//
// Algebraic reductions:
//  (1) The weight circuit is a fixed 32x32 matrix M_g per generator; the
//      per-sample part is a product state s0 = ⊗_w (cos(n_w/2), sin(n_w/2)).
//  (2) The reference's sum-normalization cancels: probs[:16]/sum then /max
//      == probs[:16]/max(probs[:16]).  So only the FIRST 16 rows of M_g are
//      ever needed -> (B x 32) @ M_g[:16]^T f32 GEMM for 4 generators.
// Core runs on V_WMMA_F32_16X16X4_F32 (32 WMMAs / 16-sample wave tile).
// Memory-bound: ~68 MB traffic -> ~3 us at 23.3 TB/s.

typedef float v2f __attribute__((ext_vector_type(2)));
typedef float v8f __attribute__((ext_vector_type(8)));

#define NQ   5
#define DIM  32     // 2^NQ
#define QD   6
#define NGEN 4
#define PAD  34     // even pad: 8B-aligned b64 LDS loads; 34n+2h mod 64 distinct except (n,h)=(15,1),(0,0)

// ds_swizzle xor-butterfly within 16-lane halves (offset[15]=0 group-of-32:
// offset = xor_mask<<10 | or_mask<<5 | and_mask; and=0x1f, or=0).
template <int IMM>
__device__ __forceinline__ float swz_max(float v) {
  float o = __int_as_float(__builtin_amdgcn_ds_swizzle(__float_as_int(v), IMM));
  return fmaxf(v, o);
}
#define SWZ_X1 0x041F
#define SWZ_X2 0x081F
#define SWZ_X4 0x101F
#define SWZ_X8 0x201F

// ---------------------------------------------------------------------------
// Kernel 1: build M_g (32x32, row-major [g][out j][in i]) from q_params.
// 128 threads: thread = (g, basis column). State kept fully in VGPRs.
// ---------------------------------------------------------------------------
__global__ __launch_bounds__(128) void qgen_build_M(const float* __restrict__ qp,
                                                    float* __restrict__ Mout) {
  const int t   = threadIdx.x;
  const int g   = t >> 5;
  const int col = t & 31;
  float st[DIM];
#pragma unroll
  for (int j = 0; j < DIM; ++j) st[j] = (j == col) ? 1.0f : 0.0f;

  for (int d = 0; d < QD; ++d) {
#pragma unroll
    for (int w = 0; w < NQ; ++w) {
      float s, c;
      sincosf(0.5f * qp[g * 30 + d * 5 + w], &s, &c);
      const int bit = 1 << (4 - w);   // wire 0 = MSB of flattened index
#pragma unroll
      for (int j = 0; j < DIM; ++j) {
        if (!(j & bit)) {
          float a0 = st[j], a1 = st[j | bit];
          st[j]       = c * a0 - s * a1;
          st[j | bit] = s * a0 + c * a1;
        }
      }
    }
#pragma unroll
    for (int w = 0; w < NQ - 1; ++w) {   // CZ chain: sign flip where both bits set
      const int b1 = 1 << (4 - w), b2 = 1 << (3 - w);
#pragma unroll
      for (int j = 0; j < DIM; ++j)
        if ((j & b1) && (j & b2)) st[j] = -st[j];
    }
  }
#pragma unroll
  for (int j = 0; j < DIM; ++j) Mout[g * 1024 + j * 32 + col] = st[j];
}

// ---------------------------------------------------------------------------
// Kernel 2: main pass. Block = 256 threads = 8 waves; each wave owns 16
// samples. Phase 1: lanes 0-15 build s0[32] into LDS. Phase 2: 32x
// v_wmma_f32_16x16x4_f32 (4 gens x 8 K-steps, N-tile = outputs 0..15 only).
// Epilogue: square, 16-lane max via ds_swizzle butterflies, out = p/max.
// ---------------------------------------------------------------------------
__global__ __launch_bounds__(256) void qgen_main(const float* __restrict__ in,
                                                 const float* __restrict__ M,
                                                 float* __restrict__ out) {
  __shared__ float sM[NGEN * 16 * PAD];    // 4 gens x first-16 rows (padded)
  __shared__ float sA[128 * PAD];          // 8 waves x 16 samples (padded)

  const int tid = threadIdx.x;
  // Stage the needed half of M (rows 0..15 per generator) into LDS.
  for (int idx = tid; idx < NGEN * 16 * DIM; idx += 256) {
    const int g = idx >> 9, rem = idx & 511;
    const int row = rem >> 5, colm = rem & 31;
    sM[(g * 16 + row) * PAD + colm] = M[g * 1024 + row * 32 + colm];
  }

  const int wave = tid >> 5;
  const int lane = tid & 31;
  const int lh   = lane >> 4;   // lane half (selects K pair / D row group)
  const int n    = lane & 15;   // N / sample index within tile
  const size_t baseSample = (size_t)blockIdx.x * 128u + (size_t)wave * 16u;

  if (lane < 16) {
    // Bilinear row-0 taps: only first 3 floats of the 4x4 patch matter.
    const float* rp = in + (baseSample + (size_t)lane) * 16u;
    const float r0 = rp[0], r1 = rp[1], r2 = rp[2];
    float ang[5] = { r0,
                     0.75f * r0 + 0.25f * r1,
                     0.25f * r0 + 0.75f * r1,
                     0.75f * r1 + 0.25f * r2,
                     0.25f * r1 + 0.75f * r2 };
    float cw[5], sw[5];
#pragma unroll
    for (int w = 0; w < 5; ++w) __sincosf(0.5f * ang[w], &sw[w], &cw[w]);
    float* dst = &sA[(wave * 16 + lane) * PAD];
#pragma unroll
    for (int i = 0; i < DIM; ++i) {
      float v = ((i >> 4) & 1) ? sw[0] : cw[0];
      v *= ((i >> 3) & 1) ? sw[1] : cw[1];
      v *= ((i >> 2) & 1) ? sw[2] : cw[2];
      v *= ((i >> 1) & 1) ? sw[3] : cw[3];
      v *= ( i       & 1) ? sw[4] : cw[4];
      dst[i] = v;
    }
  }
  __syncthreads();

  // --- WMMA phase (EXEC all-1s: B divisible by 128, no divergence here) ---
  // A layout (ISA 7.12.2, 32-bit A 16x4): lane(lh,n) holds A[n][4k+2*lh+{0,1}]
  // B layout (mirror):                    lane(lh,n) holds B[2*lh+{0,1}][n]
  v8f acc[NGEN] = {};
  const float* aRow = &sA[(wave * 16 + n) * PAD + 2 * lh];
#pragma unroll
  for (int k = 0; k < 8; ++k) {
    const v2f a = *(const v2f*)(aRow + 4 * k);
#pragma unroll
    for (int g = 0; g < NGEN; ++g) {
      const v2f b = *(const v2f*)&sM[(g * 16 + n) * PAD + 4 * k + 2 * lh];
      acc[g] = __builtin_amdgcn_wmma_f32_16x16x4_f32(
          false, a, false, b, (short)0, acc[g], false, false);
    }
  }

  // --- Epilogue: D row M = r + 8*lh -> sample = base + 8*lh + r; col = n ---
  // Output = probs[:16] / max(probs[:16]) (sum-normalization cancels exactly).
  const size_t sBase = baseSample + (size_t)lh * 8u;
#pragma unroll
  for (int g = 0; g < NGEN; ++g) {
#pragma unroll
    for (int r = 0; r < 8; ++r) {
      float p = acc[g][r];
      p = p * p;
      float mx = p;                       // max over the 16 lanes of this half
      mx = swz_max<SWZ_X1>(mx);
      mx = swz_max<SWZ_X2>(mx);
      mx = swz_max<SWZ_X4>(mx);
      mx = swz_max<SWZ_X8>(mx);
      out[(sBase + (size_t)r) * 64u + (size_t)(g * 16 + n)] = p / mx;
    }
  }
}

extern "C" void kernel_launch(void* const* d_in, const int* in_sizes, int n_in,
                              void* d_out, int out_size, void* d_ws, size_t ws_size,
                              hipStream_t stream) {
  const float* inp = (const float*)d_in[0];   // (B,1,4,4) f32
  const float* qp  = (const float*)d_in[1];   // (4,30) f32
  float* outp = (float*)d_out;                // (B,64) f32
  float* Mws  = (float*)d_ws;                 // 4*32*32 f32 = 16 KB scratch

  qgen_build_M<<<1, 128, 0, stream>>>(qp, Mws);

  const int B = in_sizes[0] / 16;             // 262144; divisible by 128
  qgen_main<<<B / 128, 256, 0, stream>>>(inp, Mws, outp);
}